// MagneticEdgeLaplacianConv_17205638988403
// MI455X (gfx1250) — compile-verified
//
#include <hip/hip_runtime.h>
#include <hip/hip_bf16.h>

typedef __attribute__((ext_vector_type(2))) float        v2f;
typedef __attribute__((ext_vector_type(8))) float        v8f;
typedef __attribute__((ext_vector_type(4))) unsigned int v4u;
typedef __attribute__((ext_vector_type(8))) int          v8i;
typedef __attribute__((ext_vector_type(4))) int          v4i;

#define E_EDGES 4096
#define N_NODES 1024
#define IN_DIM  256
#define OUT_DIM 256
#define CHALF   128   // OUT_DIM/2 complex channels

// TDM pads 4 dwords after every 256 dwords stored -> LDS row pitch 260 dwords.
#define LDS_PITCH 260

// complex64(np.exp(1j*pi*1.0))
#define PH_R (-1.0f)
#define PH_I (1.2246468e-16f)

// Per-edge incidence column values (signed_in == signed_out => B_in == B_out):
//   u_e = -a_e at row src_e,  v_e = b_e at row dst_e
//   a = dir ? phase : 1 ; b = dir ? conj(phase) : 1
__device__ __forceinline__ void edge_uv(bool dir, float& ur, float& ui,
                                        float& vr, float& vi) {
  ur = dir ?  1.0f : -1.0f;
  ui = dir ? -PH_I :  0.0f;
  vr = dir ? -1.0f :  1.0f;
  vi = dir ? -PH_I :  0.0f;
}

// ---------------------------------------------------------------- zero t ----
__global__ __launch_bounds__(256) void zero_t_kernel(float* __restrict__ t) {
  int i = blockIdx.x * blockDim.x + threadIdx.x;
  if (i < N_NODES * OUT_DIM) t[i] = 0.0f;
}

// ------------------------------------------------- h = x @ W^T via WMMA ----
// Block = one 16-row m-tile of x, staged into LDS once by the Tensor Data
// Mover (with hardware padding for bank-conflict-free column reads).
// 16 waves per block: wave w computes output n-tile w.  K=256 -> 64 WMMAs.
// A (16x4, MxK): lane l -> M=l&15, K pair base=(l>=16)?2:0  (from LDS)
// B (4x16, KxN): lane l -> N=l&15, K pair base=(l>=16)?2:0, B[k][n]=W[n][k]
// C/D (16x16):   vgpr j -> M=j+((l>=16)?8:0), N=l&15
__global__ __launch_bounds__(512) void gemm_xWt_kernel(const float* __restrict__ x,
                                                       const float* __restrict__ W,
                                                       float* __restrict__ h) {
  __shared__ float xtile[16 * LDS_PITCH];   // 16.6 KB

  const int tm   = blockIdx.x;              // 0..255  (m-tile)
  const int tn   = threadIdx.x >> 5;        // 0..15   (n-tile == wave id)
  const int lane = threadIdx.x & 31;
  const int r    = lane & 15;
  const int kg   = (lane >> 4) << 1;        // 0 or 2

  if (tn == 0) {
    // --- Tensor DMA: x[tm*16 .. tm*16+15][0..255] -> LDS (padded rows) ---
    const unsigned long long gaddr =
        (unsigned long long)(uintptr_t)(x + (size_t)tm * 16 * IN_DIM);
    const unsigned lds_off = (unsigned)(size_t)(&xtile[0]);

    // D# group 0: count=1 | lds_addr | global_addr[56:0] | type=2
    v4u g0 = { 1u,
               lds_off,
               (unsigned)gaddr,
               (unsigned)((gaddr >> 32) & 0x1FFFFFFu) | (2u << 30) };
    // D# group 1: data_size=4B(2), pad_enable, pad_interval=7(256dw),
    // pad_amount=3(4dw); tensor 256x16, tile 256x16, dim0 stride 256.
    v8i g1 = { (int)((2u << 16) | (1u << 20) | (7u << 22) | (3u << 25)),
               (int)(256u << 16),        // tensor_dim0 = 256 (bits 79:48)
               (int)(16u  << 16),        // tensor_dim1 = 16  (bits 111:80)
               (int)(256u << 16),        // tile_dim0   = 256 (bits 127:112)
               16,                       // tile_dim1   = 16  (bits 143:128)
               256,                      // tensor_dim0_stride (bits 207:160)
               0, 0 };
    v4i gz4 = { 0, 0, 0, 0 };
    v8i gz8 = { 0, 0, 0, 0, 0, 0, 0, 0 };
    __builtin_amdgcn_tensor_load_to_lds(g0, g1, gz4, gz4, gz8, 0);
    __builtin_amdgcn_s_wait_tensorcnt(0);
  }
  __syncthreads();

  const float* __restrict__ arow = &xtile[r * LDS_PITCH + kg];
  const float* __restrict__ wrow = W + (tn * 16 + r) * IN_DIM + kg;

  v8f c = {};
#pragma unroll
  for (int k = 0; k < IN_DIM / 4; ++k) {
    v2f a = *(const v2f*)(arow + k * 4);            // ds_load_b64
    v2f b = *(const v2f*)(wrow + k * 4);            // global_load_b64
    c = __builtin_amdgcn_wmma_f32_16x16x4_f32(false, a, false, b,
                                              (short)0, c, false, false);
  }

  const int mbase = tm * 16 + ((lane >> 4) ? 8 : 0);
  const int n     = tn * 16 + (lane & 15);
#pragma unroll
  for (int j = 0; j < 8; ++j) {
    h[(mbase + j) * OUT_DIM + n] = c[j];
  }
}

// ------------------------------------- deg[e] = sum_f |L[e,f]| (pair scan) --
__global__ __launch_bounds__(256) void deg_kernel(const int* __restrict__ src,
                                                  const int* __restrict__ dst,
                                                  const unsigned char* __restrict__ dir,
                                                  float* __restrict__ deg) {
  __shared__ float red[256];
  const int e   = blockIdx.x;
  const int tid = threadIdx.x;

  const int se = src[e], de = dst[e];
  float uer, uei, ver, vei;
  edge_uv(dir[e] != 0, uer, uei, ver, vei);

  float acc = 0.0f;
  for (int f = tid; f < E_EDGES; f += 256) {
    const int sf = src[f], df = dst[f];
    float ufr, ufi, vfr, vfi;
    edge_uv(dir[f] != 0, ufr, ufi, vfr, vfi);

    float Lr = 0.0f, Li = 0.0f;
    bool hit = false;
    if (se == sf) { Lr += uer * ufr + uei * ufi; Li += uer * ufi - uei * ufr; hit = true; }
    if (se == df) { Lr += uer * vfr + uei * vfi; Li += uer * vfi - uei * vfr; hit = true; }
    if (de == sf) { Lr += ver * ufr + vei * ufi; Li += ver * ufi - vei * ufr; hit = true; }
    if (de == df) { Lr += ver * vfr + vei * vfi; Li += ver * vfi - vei * vfr; hit = true; }
    if (hit) acc += sqrtf(Lr * Lr + Li * Li);
  }

  red[tid] = acc;
  __syncthreads();
  for (int s = 128; s > 0; s >>= 1) {
    if (tid < s) red[tid] += red[tid + s];
    __syncthreads();
  }
  if (tid == 0) deg[e] = red[0];
}

// ------------------------- t = B * (D^{-1/2} hc)  (atomic scatter to nodes) --
__global__ __launch_bounds__(128) void scatter_kernel(const int* __restrict__ src,
                                                      const int* __restrict__ dst,
                                                      const unsigned char* __restrict__ dir,
                                                      const float* __restrict__ h,
                                                      const float* __restrict__ deg,
                                                      float* __restrict__ t) {
  const int e = blockIdx.x;
  const int c = threadIdx.x;   // complex channel 0..127

  const float d    = deg[e];
  const float dinv = (d > 0.0f) ? (1.0f / sqrtf(d)) : 0.0f;

  const float gr = dinv * h[e * OUT_DIM + 2 * c];
  const float gi = dinv * h[e * OUT_DIM + 2 * c + 1];

  float ur, ui, vr, vi;
  edge_uv(dir[e] != 0, ur, ui, vr, vi);

  float* ts = t + src[e] * OUT_DIM + 2 * c;
  float* td = t + dst[e] * OUT_DIM + 2 * c;
  atomicAdd(ts,     ur * gr - ui * gi);
  atomicAdd(ts + 1, ur * gi + ui * gr);
  atomicAdd(td,     vr * gr - vi * gi);
  atomicAdd(td + 1, vr * gi + vi * gr);
}

// ------------------------------ y = D^{-1/2} * B^H * t  (gather per edge) ---
__global__ __launch_bounds__(128) void gather_kernel(const int* __restrict__ src,
                                                     const int* __restrict__ dst,
                                                     const unsigned char* __restrict__ dir,
                                                     const float* __restrict__ deg,
                                                     const float* __restrict__ t,
                                                     float* __restrict__ out) {
  const int e = blockIdx.x;
  const int c = threadIdx.x;

  const float d    = deg[e];
  const float dinv = (d > 0.0f) ? (1.0f / sqrtf(d)) : 0.0f;

  float ur, ui, vr, vi;
  edge_uv(dir[e] != 0, ur, ui, vr, vi);

  const float* ts = t + src[e] * OUT_DIM + 2 * c;
  const float* td = t + dst[e] * OUT_DIM + 2 * c;
  const float tr = ts[0], ti = ts[1];
  const float wr = td[0], wi = td[1];

  // conj(u)*t_src + conj(v)*t_dst
  const float yr = (ur * tr + ui * ti) + (vr * wr + vi * wi);
  const float yi = (ur * ti - ui * tr) + (vr * wi - vi * wr);

  out[e * OUT_DIM + 2 * c]     = dinv * yr;
  out[e * OUT_DIM + 2 * c + 1] = dinv * yi;
}

extern "C" void kernel_launch(void* const* d_in, const int* in_sizes, int n_in,
                              void* d_out, int out_size, void* d_ws, size_t ws_size,
                              hipStream_t stream) {
  (void)in_sizes; (void)n_in; (void)out_size; (void)ws_size;

  const float*         x   = (const float*)d_in[0];          // [E, IN]
  const float*         W   = (const float*)d_in[1];          // [OUT, IN]
  const int*           ei  = (const int*)d_in[2];            // [2, E]
  const unsigned char* dir = (const unsigned char*)d_in[3];  // [E] bool
  float*               out = (float*)d_out;                  // [E, OUT]

  float* ws  = (float*)d_ws;
  float* h   = ws;                                // E*OUT    = 1,048,576 f
  float* t   = h + (size_t)E_EDGES * OUT_DIM;     // N*OUT    =   262,144 f
  float* deg = t + (size_t)N_NODES * OUT_DIM;     // E        =     4,096 f

  const int* src = ei;
  const int* dst = ei + E_EDGES;

  zero_t_kernel<<<(N_NODES * OUT_DIM + 255) / 256, 256, 0, stream>>>(t);

  // 256 m-tiles; 16 waves/block, one n-tile per wave; x tile staged via TDM.
  gemm_xWt_kernel<<<256, 512, 0, stream>>>(x, W, h);

  deg_kernel<<<E_EDGES, 256, 0, stream>>>(src, dst, dir, deg);

  scatter_kernel<<<E_EDGES, CHALF, 0, stream>>>(src, dst, dir, h, deg, t);

  gather_kernel<<<E_EDGES, CHALF, 0, stream>>>(src, dst, dir, deg, t, out);
}